// MuleHunterGNN_39530878992899
// MI455X (gfx1250) — compile-verified
//
#include <hip/hip_runtime.h>
#include <hip/hip_bf16.h>
#include <math.h>

typedef __attribute__((ext_vector_type(2))) float v2f;
typedef __attribute__((ext_vector_type(8))) float v8f;

#define HID 64
#define HEADS 4

// ---------------------------------------------------------------------------
// WMMA f32 16x16x4 tile accumulation: C += A[rowBase:+16, :K] @ B[:K, colBase:+16]
// A row-major lda=K, B row-major ldb=OUT. Optional per-row scale on A.
// Wave32, EXEC all ones, one wave per 16x16 tile.
// ---------------------------------------------------------------------------
__device__ inline v8f acc_tile(const float* __restrict__ A, int lda, int rowBase,
                               const float* __restrict__ rowScale,
                               const float* __restrict__ B, int ldb, int colBase,
                               int K, v8f c) {
  const int lane = threadIdx.x & 31;
  const int half = lane >> 4;   // 0: K pair {0,1}, 1: K pair {2,3}
  const int lr   = lane & 15;
  const int arow = rowBase + lr;
  const float s = rowScale ? rowScale[arow] : 1.0f;
  const float* Ap = A + (size_t)arow * lda;
  const float* Bp = B + colBase + lr;
  for (int k0 = 0; k0 < K; k0 += 4) {
    const int ka = k0 + 2 * half;
    v2f a, b;
    a.x = Ap[ka] * s;
    a.y = Ap[ka + 1] * s;
    b.x = Bp[(size_t)ka * ldb];
    b.y = Bp[(size_t)(ka + 1) * ldb];
    c = __builtin_amdgcn_wmma_f32_16x16x4_f32(false, a, false, b, (short)0, c,
                                              false, false);
  }
  return c;
}

// D[rowBase:+16, colBase:+16] with fused epilogue:
//   val = (A*rowScale)@WL + (A2@WR) + bias ; BN ; ReLU ; += addIn ; store
__global__ void wmma_gemm(const float* __restrict__ A,
                          const float* __restrict__ rowScale,
                          const float* __restrict__ WL,
                          const float* __restrict__ A2,
                          const float* __restrict__ WR,
                          const float* __restrict__ bias,
                          const float* __restrict__ bn_g,
                          const float* __restrict__ bn_b,
                          const float* __restrict__ bn_m,
                          const float* __restrict__ bn_v,
                          const float* __restrict__ addIn,
                          float* __restrict__ D,
                          int K, int OUT, int doRelu) {
  const int rowBase = blockIdx.x * 16;
  const int colBase = blockIdx.y * 16;
  const int lane = threadIdx.x & 31;
  const int half = lane >> 4;
  const int lr   = lane & 15;

  v8f c = {};
  if (A)  c = acc_tile(A,  K, rowBase, rowScale, WL, OUT, colBase, K, c);
  if (A2) c = acc_tile(A2, K, rowBase, nullptr,  WR, OUT, colBase, K, c);

  const int col = colBase + lr;
  const float bval = bias ? bias[col] : 0.0f;
  float sc = 1.0f, sh = 0.0f;
  if (bn_g) {
    sc = bn_g[col] * rsqrtf(bn_v[col] + 1e-5f);
    sh = bn_b[col] - bn_m[col] * sc;
  }
  #pragma unroll
  for (int v = 0; v < 8; ++v) {
    const int row = rowBase + v + 8 * half;
    float val = c[v] + bval;
    if (bn_g)   val = val * sc + sh;
    if (doRelu) val = fmaxf(val, 0.0f);
    if (addIn)  val += addIn[(size_t)row * OUT + col];
    D[(size_t)row * OUT + col] = val;
  }
}

// ---------------------------------------------------------------------------
// Graph kernels
// ---------------------------------------------------------------------------
__global__ void k_degree(const long long* __restrict__ ei, float* __restrict__ deg,
                         int E) {
  int t = blockIdx.x * blockDim.x + threadIdx.x;
  if (t < E) atomicAdd(&deg[(int)ei[E + t]], 1.0f);
}

__global__ void k_invdeg(float* __restrict__ deg, int N) {
  int t = blockIdx.x * blockDim.x + threadIdx.x;
  if (t < N) deg[t] = 1.0f / fmaxf(deg[t], 1.0f);
}

// acc[tgt, 0:64] += X[src, 0:64], float4 per thread (E*16 threads)
__global__ void k_scatter64(const long long* __restrict__ ei,
                            const float* __restrict__ X,
                            float* __restrict__ acc, int E) {
  int t = blockIdx.x * blockDim.x + threadIdx.x;
  int e = t >> 4;
  if (e >= E) return;
  int q = t & 15;
  int src = (int)ei[e];
  int tgt = (int)ei[E + e];
  const float4 v = *(const float4*)(X + (size_t)src * 64 + q * 4);
  float* p = acc + (size_t)tgt * 64 + q * 4;
  atomicAdd(p + 0, v.x);
  atomicAdd(p + 1, v.y);
  atomicAdd(p + 2, v.z);
  atomicAdd(p + 3, v.w);
}

// per (node, head): attention logits al_s, al_d
__global__ void k_attlogits(const float* __restrict__ hgat,
                            const float* __restrict__ asrc,
                            const float* __restrict__ adst,
                            float* __restrict__ als, float* __restrict__ ald,
                            int N) {
  int t = blockIdx.x * blockDim.x + threadIdx.x;
  if (t >= N * HEADS) return;
  int n = t >> 2, h = t & 3;
  const float* hp = hgat + (size_t)n * (HEADS * HID) + h * HID;
  const float* ap = asrc + h * HID;
  const float* dp = adst + h * HID;
  float s = 0.0f, d = 0.0f;
  for (int c = 0; c < HID; ++c) {
    float v = hp[c];
    s += v * ap[c];
    d += v * dp[c];
  }
  als[t] = s;
  ald[t] = d;
}

__device__ inline unsigned fenc(float v) {
  unsigned u = __float_as_uint(v);
  return (u & 0x80000000u) ? ~u : (u | 0x80000000u);
}
__device__ inline float fdec(unsigned u) {
  return (u & 0x80000000u) ? __uint_as_float(u & 0x7fffffffu)
                           : __uint_as_float(~u);
}

__device__ inline float leaky02(float x) { return x > 0.0f ? x : 0.2f * x; }

// per (edge, head): running max of leaky-relu logits (flipped-uint atomicMax)
__global__ void k_edge_max(const long long* __restrict__ ei,
                           const float* __restrict__ als,
                           const float* __restrict__ ald,
                           unsigned* __restrict__ emax, int E) {
  int t = blockIdx.x * blockDim.x + threadIdx.x;
  if (t >= E * HEADS) return;
  int e = t >> 2, h = t & 3;
  int src = (int)ei[e];
  int tgt = (int)ei[E + e];
  float le = leaky02(als[src * HEADS + h] + ald[tgt * HEADS + h]);
  atomicMax(&emax[tgt * HEADS + h], fenc(le));
}

// per (edge, head): softmax denominator
__global__ void k_edge_den(const long long* __restrict__ ei,
                           const float* __restrict__ als,
                           const float* __restrict__ ald,
                           const unsigned* __restrict__ emax,
                           float* __restrict__ den, int E) {
  int t = blockIdx.x * blockDim.x + threadIdx.x;
  if (t >= E * HEADS) return;
  int e = t >> 2, h = t & 3;
  int src = (int)ei[e];
  int tgt = (int)ei[E + e];
  float le = leaky02(als[src * HEADS + h] + ald[tgt * HEADS + h]);
  unsigned u = emax[tgt * HEADS + h];
  float em = u ? fdec(u) : 0.0f;
  atomicAdd(&den[tgt * HEADS + h], expf(le - em));
}

// per (edge, 64 channels): acc64[tgt,c] += 0.25 * h[src,h,c] * alpha_{e,h}
__global__ void k_edge_accum(const long long* __restrict__ ei,
                             const float* __restrict__ als,
                             const float* __restrict__ ald,
                             const unsigned* __restrict__ emax,
                             const float* __restrict__ den,
                             const float* __restrict__ hgat,
                             float* __restrict__ acc, int E) {
  int t = blockIdx.x * blockDim.x + threadIdx.x;
  int e = t >> 6;
  if (e >= E) return;
  int r = t & 63;
  int h = r >> 4;       // head
  int q = r & 15;       // float4 chunk within 64 channels
  int src = (int)ei[e];
  int tgt = (int)ei[E + e];
  float le = leaky02(als[src * HEADS + h] + ald[tgt * HEADS + h]);
  unsigned u = emax[tgt * HEADS + h];
  float em = u ? fdec(u) : 0.0f;
  float alpha = expf(le - em) / (den[tgt * HEADS + h] + 1e-16f);
  float w = 0.25f * alpha;  // head mean folded in
  const float4 hv =
      *(const float4*)(hgat + (size_t)src * (HEADS * HID) + h * HID + q * 4);
  float* p = acc + (size_t)tgt * HID + q * 4;
  atomicAdd(p + 0, w * hv.x);
  atomicAdd(p + 1, w * hv.y);
  atomicAdd(p + 2, w * hv.z);
  atomicAdd(p + 3, w * hv.w);
}

// per (node, channel): + bias, BN2, ReLU
__global__ void k_gat_final(const float* __restrict__ acc,
                            const float* __restrict__ gat_b,
                            const float* __restrict__ g,
                            const float* __restrict__ b,
                            const float* __restrict__ m,
                            const float* __restrict__ v,
                            float* __restrict__ h2, int N) {
  int t = blockIdx.x * blockDim.x + threadIdx.x;
  if (t >= N * HID) return;
  int c = t & 63;
  float sc = g[c] * rsqrtf(v[c] + 1e-5f);
  float sh = b[c] - m[c] * sc;
  float val = acc[t] + gat_b[c];
  h2[t] = fmaxf(val * sc + sh, 0.0f);
}

// per node: relu(emb@W1+b1)@W2+b2 -> log_softmax(2)
__global__ void k_cls(const float* __restrict__ emb,
                      const float* __restrict__ w1, const float* __restrict__ b1,
                      const float* __restrict__ w2, const float* __restrict__ b2,
                      float* __restrict__ out, int N) {
  int n = blockIdx.x * blockDim.x + threadIdx.x;
  if (n >= N) return;
  float e[32];
  #pragma unroll
  for (int i = 0; i < 32; ++i) e[i] = emb[(size_t)n * 32 + i];
  float z0 = b2[0], z1 = b2[1];
  for (int j = 0; j < 32; ++j) {
    float a = b1[j];
    #pragma unroll
    for (int i = 0; i < 32; ++i) a += e[i] * w1[i * 32 + j];
    a = fmaxf(a, 0.0f);
    z0 += a * w2[j * 2 + 0];
    z1 += a * w2[j * 2 + 1];
  }
  float mx = fmaxf(z0, z1);
  float l = logf(expf(z0 - mx) + expf(z1 - mx)) + mx;
  out[(size_t)n * 2 + 0] = z0 - l;
  out[(size_t)n * 2 + 1] = z1 - l;
}

// ---------------------------------------------------------------------------
extern "C" void kernel_launch(void* const* d_in, const int* in_sizes, int n_in,
                              void* d_out, int out_size, void* d_ws, size_t ws_size,
                              hipStream_t stream) {
  const float*     x       = (const float*)d_in[0];
  const long long* ei      = (const long long*)d_in[1];
  const float* skip_w = (const float*)d_in[2];
  const float* skip_b = (const float*)d_in[3];
  const float* c1_wl  = (const float*)d_in[4];
  const float* c1_b   = (const float*)d_in[5];
  const float* c1_wr  = (const float*)d_in[6];
  const float* bn1_g  = (const float*)d_in[7];
  const float* bn1_b  = (const float*)d_in[8];
  const float* bn1_m  = (const float*)d_in[9];
  const float* bn1_v  = (const float*)d_in[10];
  const float* gat_w  = (const float*)d_in[11];
  const float* a_src  = (const float*)d_in[12];
  const float* a_dst  = (const float*)d_in[13];
  const float* gat_b  = (const float*)d_in[14];
  const float* bn2_g  = (const float*)d_in[15];
  const float* bn2_b  = (const float*)d_in[16];
  const float* bn2_m  = (const float*)d_in[17];
  const float* bn2_v  = (const float*)d_in[18];
  const float* c3_wl  = (const float*)d_in[19];
  const float* c3_b   = (const float*)d_in[20];
  const float* c3_wr  = (const float*)d_in[21];
  const float* bn3_g  = (const float*)d_in[22];
  const float* bn3_b  = (const float*)d_in[23];
  const float* bn3_m  = (const float*)d_in[24];
  const float* bn3_v  = (const float*)d_in[25];
  const float* cls_w1 = (const float*)d_in[26];
  const float* cls_b1 = (const float*)d_in[27];
  const float* cls_w2 = (const float*)d_in[28];
  const float* cls_b2 = (const float*)d_in[29];
  float* out = (float*)d_out;

  const int N = in_sizes[0] / 64;      // 50000
  const int E = in_sizes[1] / 2;       // 800000

  // workspace layout (all offsets multiples of 16 floats)
  float* W = (float*)d_ws;
  size_t o = 0;
  float*    deg   = W + o; o += (size_t)N;            // -> invdeg
  float*    agg   = W + o; o += (size_t)N * 64;       // agg1 / gat acc64 / agg3
  float*    h1    = W + o; o += (size_t)N * 64;       // h1 -> h2
  float*    hgat  = W + o; o += (size_t)N * 256;
  float*    als   = W + o; o += (size_t)N * 4;
  float*    ald   = W + o; o += (size_t)N * 4;
  float*    den   = W + o; o += (size_t)N * 4;
  unsigned* emax  = (unsigned*)(W + o); o += (size_t)N * 4;
  float*    ident = W + o; o += (size_t)N * 32;       // identity -> emb

  const int TPB = 256;
  const int rowTiles = N / 16;   // 3125 (N divisible by 16)

  hipMemsetAsync(deg,  0, (size_t)N * sizeof(float), stream);
  hipMemsetAsync(agg,  0, (size_t)N * 64 * sizeof(float), stream);
  hipMemsetAsync(den,  0, (size_t)N * 4 * sizeof(float), stream);
  hipMemsetAsync(emax, 0, (size_t)N * 4 * sizeof(unsigned), stream);

  // degree + mean aggregation of x
  k_degree<<<(E + TPB - 1) / TPB, TPB, 0, stream>>>(ei, deg, E);
  k_scatter64<<<((size_t)E * 16 + TPB - 1) / TPB, TPB, 0, stream>>>(ei, x, agg, E);
  k_invdeg<<<(N + TPB - 1) / TPB, TPB, 0, stream>>>(deg, N);

  // identity = x @ skip_w + skip_b             [N,32]
  wmma_gemm<<<dim3(rowTiles, 2), 32, 0, stream>>>(
      x, nullptr, skip_w, nullptr, nullptr, skip_b,
      nullptr, nullptr, nullptr, nullptr, nullptr, ident, 64, 32, 0);

  // h1 = relu(bn1( (agg*invdeg)@c1_wl + c1_b + x@c1_wr ))   [N,64]
  wmma_gemm<<<dim3(rowTiles, 4), 32, 0, stream>>>(
      agg, deg, c1_wl, x, c1_wr, c1_b,
      bn1_g, bn1_b, bn1_m, bn1_v, nullptr, h1, 64, 64, 1);

  // hgat = h1 @ gat_w                           [N,256]
  wmma_gemm<<<dim3(rowTiles, 16), 32, 0, stream>>>(
      h1, nullptr, gat_w, nullptr, nullptr, nullptr,
      nullptr, nullptr, nullptr, nullptr, nullptr, hgat, 64, 256, 0);

  // attention
  k_attlogits<<<((size_t)N * 4 + TPB - 1) / TPB, TPB, 0, stream>>>(
      hgat, a_src, a_dst, als, ald, N);
  k_edge_max<<<((size_t)E * 4 + TPB - 1) / TPB, TPB, 0, stream>>>(
      ei, als, ald, emax, E);
  k_edge_den<<<((size_t)E * 4 + TPB - 1) / TPB, TPB, 0, stream>>>(
      ei, als, ald, emax, den, E);

  hipMemsetAsync(agg, 0, (size_t)N * 64 * sizeof(float), stream);  // acc64
  k_edge_accum<<<((size_t)E * 64 + TPB - 1) / TPB, TPB, 0, stream>>>(
      ei, als, ald, emax, den, hgat, agg, E);
  k_gat_final<<<((size_t)N * 64 + TPB - 1) / TPB, TPB, 0, stream>>>(
      agg, gat_b, bn2_g, bn2_b, bn2_m, bn2_v, h1, N);   // h1 now holds h2

  // SAGE-3: mean aggregate h2, then fused GEMM + BN3 + ReLU + residual
  hipMemsetAsync(agg, 0, (size_t)N * 64 * sizeof(float), stream);
  k_scatter64<<<((size_t)E * 16 + TPB - 1) / TPB, TPB, 0, stream>>>(ei, h1, agg, E);

  // emb = relu(bn3( (agg*invdeg)@c3_wl + c3_b + h2@c3_wr )) + identity  [N,32]
  wmma_gemm<<<dim3(rowTiles, 2), 32, 0, stream>>>(
      agg, deg, c3_wl, h1, c3_wr, c3_b,
      bn3_g, bn3_b, bn3_m, bn3_v, ident, ident, 64, 32, 1);

  // classifier + log_softmax
  k_cls<<<(N + 127) / 128, 128, 0, stream>>>(
      ident, cls_w1, cls_b1, cls_w2, cls_b2, out, N);
}